// CorrelationLayer_35493609734313
// MI455X (gfx1250) — compile-verified
//
#include <hip/hip_runtime.h>
#include <hip/hip_bf16.h>

typedef __attribute__((ext_vector_type(2))) float v2f;
typedef __attribute__((ext_vector_type(8))) float v8f;
typedef __attribute__((ext_vector_type(4))) int   v4i;

#define CC   128
#define HH   48
#define WW   64
#define HW   (HH * WW)     // 3072
#define ND   21            // displacements per axis
#define PAD  20

#define AS1 __attribute__((address_space(1)))
#define AS3 __attribute__((address_space(3)))

#if __has_builtin(__builtin_amdgcn_global_load_async_to_lds_b128) && \
    __has_builtin(__builtin_amdgcn_s_wait_asynccnt)
#define USE_ASYNC_LDS 1
#else
#define USE_ASYNC_LDS 0
#endif

// One block per (dy, i): stages f1 row i and f2 row (i+dy-20) into LDS
// (async global->LDS on CDNA5), then 4 waves each compute a 16-pixel
// (same-parity) j-tile against 3 shifted 16-column m-tiles of f2 via
// chained V_WMMA_F32_16X16X4_F32. A fragments are cached in VGPRs across
// the 3 B-tiles.
__global__ __launch_bounds__(128)
void corr_wmma_kernel(const float* __restrict__ f1,
                      const float* __restrict__ f2,
                      float* __restrict__ out) {
    __shared__ __align__(16) float ldsA[CC * WW];   // f1[c][j], 32 KB
    __shared__ __align__(16) float ldsB[CC * WW];   // f2[c][m], 32 KB

    const int dy = blockIdx.x;        // 0..20
    const int i  = blockIdx.y;        // 0..47
    const int r  = i + dy - PAD;      // f2 source row (may be out of range)
    const bool r_ok = (r >= 0) && (r < HH);

    // ---- Stage f1 row i and f2 row r into LDS ----
    {
        const float4* src1 = reinterpret_cast<const float4*>(f1 + (size_t)i * WW);
        const float4* src2 = r_ok
            ? reinterpret_cast<const float4*>(f2 + (size_t)r * WW)
            : nullptr;
        float4* dA = reinterpret_cast<float4*>(ldsA);
        float4* dB = reinterpret_cast<float4*>(ldsB);
        const float4 zero4 = make_float4(0.f, 0.f, 0.f, 0.f);

#if USE_ASYNC_LDS
#pragma unroll
        for (int it = 0; it < 16; ++it) {
            int idx = threadIdx.x + it * 128;      // 0..2047 float4 slots
            int c   = idx >> 4;                    // channel
            int v   = idx & 15;                    // float4 within row
            __builtin_amdgcn_global_load_async_to_lds_b128(
                (AS1 v4i*)(src1 + c * (HW / 4) + v),
                (AS3 v4i*)(dA + c * 16 + v), 0, 0);
            if (r_ok) {
                __builtin_amdgcn_global_load_async_to_lds_b128(
                    (AS1 v4i*)(src2 + c * (HW / 4) + v),
                    (AS3 v4i*)(dB + c * 16 + v), 0, 0);
            } else {
                dB[c * 16 + v] = zero4;            // zero-fill padded row
            }
        }
        __builtin_amdgcn_s_wait_asynccnt(0);
#else
#pragma unroll
        for (int it = 0; it < 16; ++it) {
            int idx = threadIdx.x + it * 128;
            int c   = idx >> 4;
            int v   = idx & 15;
            dA[c * 16 + v] = src1[c * (HW / 4) + v];
            dB[c * 16 + v] = r_ok ? src2[c * (HW / 4) + v] : zero4;
        }
#endif
    }
    __syncthreads();

    // ---- Wave/lane decomposition ----
    const int lane = threadIdx.x & 31;
    const int wv   = threadIdx.x >> 5;     // 0..3
    const int half = lane >> 4;            // K-half selector (WMMA A/B layout)
    const int n    = lane & 15;            // M (for A rows) / N (for B cols)

    // A-tile: 16 same-parity pixels j = jbase + 2*M, jbase in {0,1,32,33}
    const int jbase = ((wv >> 1) << 5) + (wv & 1);
    const int jl    = jbase + 2 * n;       // this lane's A row pixel

    // Cache all 32 A fragments (K=128) in VGPRs: reused by all 3 B-tiles.
    v2f a[32];
#pragma unroll
    for (int cb = 0; cb < 32; ++cb) {
        const int cA = cb * 4 + 2 * half;
        a[cb].x = ldsA[cA * WW + jl];
        a[cb].y = ldsA[(cA + 1) * WW + jl];
    }

#pragma unroll
    for (int bt = 0; bt < 3; ++bt) {
        // B-tile columns: m = jbase - 20 + 32*bt + 2*N
        const int   m    = jbase - PAD + 32 * bt + 2 * n;
        const int   mc   = m < 0 ? 0 : (m > WW - 1 ? WW - 1 : m);
        const float mask = (m == mc) ? 1.0f : 0.0f;   // zero pad region

        v8f acc = {};
#pragma unroll
        for (int cb = 0; cb < 32; ++cb) {             // K = 128 in steps of 4
            const int cA = cb * 4 + 2 * half;
            v2f b;
            b.x = ldsB[cA * WW + mc] * mask;
            b.y = ldsB[(cA + 1) * WW + mc] * mask;
            acc = __builtin_amdgcn_wmma_f32_16x16x4_f32(
                /*neg_a=*/false, a[cb], /*neg_b=*/false, b,
                /*c_mod=*/(short)0, acc,
                /*reuse_a=*/false, /*reuse_b=*/false);
        }

        // D layout: element v of this lane is G[j = jbase+2*(v+8*half), m]
        // offset m - j = -20 + 32*bt + 2*(n - M)  =>  dx = 16*bt + n - M
        const int dxb = 16 * bt + n;
#pragma unroll
        for (int v = 0; v < 8; ++v) {
            const int M  = v + 8 * half;
            const int dx = dxb - M;
            if (dx >= 0 && dx <= PAD) {
                const int j = jbase + 2 * M;
                out[((size_t)(dy * ND + dx) * HH + i) * WW + j] = acc[v];
            }
        }
    }
}

extern "C" void kernel_launch(void* const* d_in, const int* in_sizes, int n_in,
                              void* d_out, int out_size, void* d_ws, size_t ws_size,
                              hipStream_t stream) {
    (void)in_sizes; (void)n_in; (void)d_ws; (void)ws_size; (void)out_size;
    const float* f1 = (const float*)d_in[0];
    const float* f2 = (const float*)d_in[1];
    float* out = (float*)d_out;
    dim3 grid(ND, HH);   // (dy, row)
    corr_wmma_kernel<<<grid, 128, 0, stream>>>(f1, f2, out);
}